// POIEncoderGCN_64020782514422
// MI455X (gfx1250) — compile-verified
//
#include <hip/hip_runtime.h>

typedef __attribute__((ext_vector_type(2))) float v2f;
typedef __attribute__((ext_vector_type(8))) float v8f;

#define F 128
#define LDA 132  // padded LDS stride: bank = (4l+k)%64 -> conflict-free, 16B-aligned rows

// ---------------- degree / normalization ----------------
__global__ void k_zero(float* __restrict__ deg, int V) {
    int i = blockIdx.x * 256 + threadIdx.x;
    if (i < V) deg[i] = 0.0f;
}

__global__ void k_deg(const long long* __restrict__ dst,
                      const float* __restrict__ ew,
                      float* __restrict__ deg, int E) {
    int e = blockIdx.x * 256 + threadIdx.x;
    if (e < E) atomicAdd(&deg[dst[e]], ew[e]);
}

__global__ void k_dinv(float* __restrict__ deg, int V) {
    int i = blockIdx.x * 256 + threadIdx.x;
    if (i < V) deg[i] = rsqrtf(deg[i] + 1.0f);  // +1 = self-loop weight
}

// ---------------- fp32 WMMA GEMM:  y = dinv * (X @ W), acc = y ----------------
// Block: 256 threads = 8 waves, computes a 32-row x 128-col tile.
// Each wave owns one 16-col N-tile and TWO 16-row M-tiles (independent
// accumulators -> 2x WMMA ILP; one B fetch feeds both tiles).
__global__ __launch_bounds__(256)
void k_gemm(const float* __restrict__ X, const float* __restrict__ W,
            const float* __restrict__ dinv,
            float* __restrict__ y, float* __restrict__ acc, int V) {
    __shared__ float lA[32 * LDA];
    const int tid  = threadIdx.x;
    const int row0 = blockIdx.x * 32;

    // stage 32x128 A tile into LDS: 4096 floats, 16 per thread (float4 x4)
    {
        int r = tid >> 3;          // 0..31 (8 threads per row)
        int c = (tid & 7) * 16;    // 0,16,...,112
        int rr = row0 + r; if (rr >= V) rr = V - 1;
        const float* s = X + (size_t)rr * F + c;
        float* d = lA + r * LDA + c;   // 16B aligned (132*4 % 16 == 0, c%4==0)
        #pragma unroll
        for (int j = 0; j < 16; j += 4)
            *(float4*)(d + j) = *(const float4*)(s + j);
    }
    __syncthreads();

    const int lane = tid & 31;
    const int wave = tid >> 5;
    const int half = lane >> 4;   // 0: K pair {0,1}; 1: K pair {2,3}
    const int l    = lane & 15;
    const int n0   = wave * 16;

    v8f c0 = {};
    v8f c1 = {};
    #pragma unroll 4
    for (int k = 0; k < F; k += 4) {
        // B 4x16 layout: lanes 0-15 -> K=k..k+1 @ N=n0+l ; lanes 16-31 -> K=k+2..k+3
        const float* bp = W + (size_t)(k + half * 2) * F + n0 + l;
        v2f b = { bp[0], bp[F] };
        // A 16x4 layout, M-tile 0 and M-tile 1
        const float* ap0 = lA + l * LDA + k + half * 2;
        const float* ap1 = ap0 + 16 * LDA;
        v2f a0 = { ap0[0], ap0[1] };
        v2f a1 = { ap1[0], ap1[1] };
        c0 = __builtin_amdgcn_wmma_f32_16x16x4_f32(false, a0, false, b,
                                                   (short)0, c0, false, false);
        c1 = __builtin_amdgcn_wmma_f32_16x16x4_f32(false, a1, false, b,
                                                   (short)0, c1, false, false);
    }

    // C layout: VGPR g, lanes 0-15 -> M=g ; lanes 16-31 -> M=g+8 ; N = l
    if (row0 + 32 <= V) {          // uniform fast path: no per-row guards
        #pragma unroll
        for (int g = 0; g < 8; ++g) {
            int r0 = row0 + g + half * 8;
            int r1 = r0 + 16;
            float v0 = c0[g] * dinv[r0];
            float v1 = c1[g] * dinv[r1];
            size_t o0 = (size_t)r0 * F + n0 + l;
            size_t o1 = o0 + (size_t)16 * F;
            y[o0] = v0;  acc[o0] = v0;   // self-loop term seeds accumulator
            y[o1] = v1;  acc[o1] = v1;
        }
    } else {
        #pragma unroll
        for (int g = 0; g < 8; ++g) {
            int r0 = row0 + g + half * 8;
            int r1 = r0 + 16;
            if (r0 < V) {
                float v0 = c0[g] * dinv[r0];
                size_t o0 = (size_t)r0 * F + n0 + l;
                y[o0] = v0;  acc[o0] = v0;
            }
            if (r1 < V) {
                float v1 = c1[g] * dinv[r1];
                size_t o1 = (size_t)r1 * F + n0 + l;
                y[o1] = v1;  acc[o1] = v1;
            }
        }
    }
}

// ---------------- edge aggregation: acc[dst] += w * y[src] ----------------
// One wave per edge; each lane handles a float4 feature chunk (512B/edge).
__global__ __launch_bounds__(256)
void k_scatter(const long long* __restrict__ src, const long long* __restrict__ dst,
               const float* __restrict__ ew, const float* __restrict__ y,
               float* __restrict__ acc, int E) {
    int idx = blockIdx.x * 256 + threadIdx.x;
    int e   = idx >> 5;
    if (e >= E) return;
    int fg = (idx & 31) * 4;
    long long s = src[e];
    long long d = dst[e];
    float w = ew[e];
    const float4 v = *(const float4*)(y + (size_t)s * F + fg);
    float* a = acc + (size_t)d * F + fg;
    atomicAdd(a + 0, w * v.x);
    atomicAdd(a + 1, w * v.y);
    atomicAdd(a + 2, w * v.z);
    atomicAdd(a + 3, w * v.w);
}

// ---------------- layer-1 epilogue: h = LN(relu(dinv*acc + b1)) ----------------
// One wave per node; 4 features per lane; wave32 shfl_xor reduction.
__global__ __launch_bounds__(256)
void k_post1(const float* __restrict__ acc, const float* __restrict__ dinv,
             const float* __restrict__ b1, const float* __restrict__ g1,
             const float* __restrict__ be1, float* __restrict__ h, int V) {
    int wave = threadIdx.x >> 5;
    int lane = threadIdx.x & 31;
    int node = blockIdx.x * 8 + wave;
    if (node >= V) return;

    float di = dinv[node];
    int f0 = lane * 4;
    size_t base = (size_t)node * F + f0;
    const float4 v = *(const float4*)(acc + base);

    float r0 = fmaxf(di * v.x + b1[f0 + 0], 0.0f);
    float r1 = fmaxf(di * v.y + b1[f0 + 1], 0.0f);
    float r2 = fmaxf(di * v.z + b1[f0 + 2], 0.0f);
    float r3 = fmaxf(di * v.w + b1[f0 + 3], 0.0f);

    float s  = r0 + r1 + r2 + r3;
    float sq = r0 * r0 + r1 * r1 + r2 * r2 + r3 * r3;
    #pragma unroll
    for (int off = 16; off > 0; off >>= 1) {
        s  += __shfl_xor(s,  off, 32);
        sq += __shfl_xor(sq, off, 32);
    }
    float mean = s * (1.0f / 128.0f);
    float var  = sq * (1.0f / 128.0f) - mean * mean;
    float inv  = rsqrtf(var + 1e-5f);

    float4 o;
    o.x = (r0 - mean) * inv * g1[f0 + 0] + be1[f0 + 0];
    o.y = (r1 - mean) * inv * g1[f0 + 1] + be1[f0 + 1];
    o.z = (r2 - mean) * inv * g1[f0 + 2] + be1[f0 + 2];
    o.w = (r3 - mean) * inv * g1[f0 + 3] + be1[f0 + 3];
    *(float4*)(h + base) = o;
}

// ---------------- layer-2 epilogue: out = dinv*acc + b2 ----------------
__global__ __launch_bounds__(256)
void k_post2(const float* __restrict__ acc, const float* __restrict__ dinv,
             const float* __restrict__ b2, float* __restrict__ out, int V) {
    int i = blockIdx.x * 256 + threadIdx.x;
    if (i < V * F) {
        int node = i >> 7;
        int f = i & (F - 1);
        out[i] = dinv[node] * acc[i] + b2[f];
    }
}

extern "C" void kernel_launch(void* const* d_in, const int* in_sizes, int n_in,
                              void* d_out, int out_size, void* d_ws, size_t ws_size,
                              hipStream_t stream) {
    const float* x   = (const float*)d_in[0];
    const float* ew  = (const float*)d_in[1];
    const float* W1  = (const float*)d_in[2];
    const float* b1  = (const float*)d_in[3];
    const float* g1  = (const float*)d_in[4];
    const float* be1 = (const float*)d_in[5];
    const float* W2  = (const float*)d_in[6];
    const float* b2  = (const float*)d_in[7];
    const long long* ei = (const long long*)d_in[8];

    const int V = in_sizes[0] / F;
    const int E = in_sizes[1];
    const long long* srcIdx = ei;       // edge_index[0]
    const long long* dstIdx = ei + E;   // edge_index[1]

    float* ws   = (float*)d_ws;
    float* dinv = ws;                               // V floats
    float* bufA = ws + (1 << 17);                   // V*F (y1, later h)
    float* bufB = bufA + (size_t)V * F;             // V*F (acc1, later y2)
    float* bufC = bufB + (size_t)V * F;             // V*F (acc2)

    const int gV   = (V + 255) / 256;
    const int gE   = (E + 255) / 256;
    const int gEF  = (int)(((long long)E * 32 + 255) / 256);
    const int gT   = (V + 31) / 32;
    const int gN   = (V + 7) / 8;
    const int gVF  = (V * F + 255) / 256;

    // normalization coefficients
    k_zero<<<gV, 256, 0, stream>>>(dinv, V);
    k_deg <<<gE, 256, 0, stream>>>(dstIdx, ew, dinv, E);
    k_dinv<<<gV, 256, 0, stream>>>(dinv, V);

    // layer 1: y1 = dinv*(x@W1) -> bufA, acc1 init = y1 -> bufB
    k_gemm   <<<gT,  256, 0, stream>>>(x, W1, dinv, bufA, bufB, V);
    k_scatter<<<gEF, 256, 0, stream>>>(srcIdx, dstIdx, ew, bufA, bufB, E);
    k_post1  <<<gN,  256, 0, stream>>>(bufB, dinv, b1, g1, be1, bufA, V);  // h -> bufA

    // layer 2: y2 = dinv*(h@W2) -> bufB, acc2 init = y2 -> bufC
    k_gemm   <<<gT,  256, 0, stream>>>(bufA, W2, dinv, bufB, bufC, V);
    k_scatter<<<gEF, 256, 0, stream>>>(srcIdx, dstIdx, ew, bufB, bufC, E);
    k_post2  <<<gVF, 256, 0, stream>>>(bufC, dinv, b2, (float*)d_out, V);
}